// BIMM2D_test_11441792876621
// MI455X (gfx1250) — compile-verified
//
#include <hip/hip_runtime.h>
#include <hip/hip_bf16.h>
#include <math.h>
#include <stdint.h>

#define LOG2E_F 1.4426950408889634f
#define LN2_F   0.6931471805599453f
#define PI_F    3.14159265358979323846f
// log(gamma(1.5)) = 0.5*log(pi) - log(2)
#define LOG_GAMMA_3_2 (-0.12078223763524522f)

// hardware transcendentals: v_exp_f32 = 2^x, v_log_f32 = log2(x)
__device__ __forceinline__ float fexp2(float x) { return __builtin_amdgcn_exp2f(x); }
__device__ __forceinline__ float flog(float x)  { return __builtin_amdgcn_logf(x) * LN2_F; }

typedef float v8f __attribute__((ext_vector_type(8)));
typedef float v2f __attribute__((ext_vector_type(2)));

__device__ __forceinline__ float sum8(v8f d) {
  return ((d[0] + d[1]) + (d[2] + d[3])) + ((d[4] + d[5]) + (d[6] + d[7]));
}

// Wave32 sum-reduction via V_WMMA_F32_16X16X4_F32 with B == ones (see round-1 notes).
__device__ __forceinline__ float wave_reduce32(float x) {
#if __has_builtin(__builtin_amdgcn_wmma_f32_16x16x4_f32)
  v2f a; a.x = x;   a.y = 0.0f;
  v2f b; b.x = 1.0f; b.y = 1.0f;
  v8f c = {0.f, 0.f, 0.f, 0.f, 0.f, 0.f, 0.f, 0.f};
  v8f d = __builtin_amdgcn_wmma_f32_16x16x4_f32(false, a, false, b, (short)0, c, false, false);
  float s = sum8(d);
  s += __shfl_xor(s, 16, 32);
  return s;
#else
  for (int off = 16; off; off >>= 1) x += __shfl_xor(x, off, 32);
  return x;
#endif
}

// ---- workspace / LDS layout (dwords) ----
//  [0..63]  params: 0 inv_sn, 1 inv_sigeff, 2 inv_sn2, 3 K_v, 4 A_int, 5 -log(N),
//           [6..13] I[k], 14 Cc = -h (u^2 coeff, log2 domain), [16..31] log_w[c]
//  [64 .. 64+3*PN)  tables (SoA, log2 domain): c2 | B | zcl      (PN = n_int*N)
//  [4096 ..] per-block partial sums
#define MAX_PN   768
#define SMEM_DW  (64 + 3 * MAX_PN)
#define PART_OFF 4096

__global__ void bimm_setup_kernel(const float* __restrict__ eps,
                                  const float* __restrict__ I,
                                  const float* __restrict__ W,
                                  const float* sb_p, const float* sn_p,
                                  const float* d_p, const float* r_p,
                                  float* __restrict__ ws,
                                  int N, int n_int, int n_phases, int ncomp) {
  const float sigma_b = sb_p[0], sigma_n = sn_p[0], dd = d_p[0], r = r_p[0];
  const float rho = tanhf(r);
  const float om = 1.0f - rho;
  const float sn2 = sigma_n * sigma_n * om;
  const float sig_eff = sigma_n * sqrtf(om);
  const float h = 0.5f * LOG2E_F / (sigma_n * sigma_n);   // log2-domain 0.5/sn^2
  const int tid = threadIdx.x;

  if (tid == 0) {
    ws[0] = 1.0f / sigma_n;
    ws[1] = 1.0f / sig_eff;
    ws[2] = 1.0f / sn2;
    ws[3] = -2.f * logf(sig_eff) - 0.5f * logf(PI_F) + 0.5f * logf(sn2)
            - logf(sigma_n) - 0.5f * logf(2.f * PI_F);          // K_v
    ws[4] = logf(2.f) - LOG_GAMMA_3_2 - 3.f * logf(sig_eff)
            - logf(sigma_n) - 0.5f * logf(2.f * PI_F);          // A_int
    ws[5] = -logf((float)N);
    for (int k = 0; k < n_phases && k < 8; k++) ws[6 + k] = I[k];
    ws[14] = -h;                                                 // Cc
    float mx = -3.4e38f;
    for (int c = 0; c < ncomp; c++) mx = fmaxf(mx, W[c]);
    float s = 0.f;
    for (int c = 0; c < ncomp; c++) s += expf(W[c] - mx);
    const float lse = mx + logf(s);
    for (int c = 0; c < ncomp && c < 16; c++) ws[16 + c] = W[c] - lse;
  }

  const int PN = n_int * N;
  float* tC2  = ws + 64;
  float* tB   = ws + 64 + PN;
  float* tZcl = ws + 64 + 2 * PN;
  const float inv_s2sb = 1.0f / (1.41421356237309515f * sigma_b);
  const float gnorm = 1.0f / sqrtf(2.f * PI_F * sigma_b * sigma_b);
  for (int idx = tid; idx < PN; idx += blockDim.x) {
    const int p = idx / N;
    int ia = 0, ib = 1, cnt = p;                // linear -> triu pair
    for (int a = 0; a < n_phases - 1; a++) {
      int row = n_phases - 1 - a;
      if (cnt < row) { ia = a; ib = a + 1 + cnt; break; }
      cnt -= row;
    }
    const float Ia = I[ia], Ib = I[ib];
    const float e  = eps[idx];
    const float ux = e * 2.f * dd * sigma_b - dd * sigma_b;
    const float arg = ux * inv_s2sb;            // == erfinv-folded argument
    const float In  = 0.5f * (erff(arg) + 1.f) * (Ib - Ia) + Ia;
    const float G   = (Ib - Ia) * gnorm * expf(-arg * arg);
    const float cn  = -logf(G) - G * G / sn2;   // natural-log constant term
    // exponent(n,m) = c2 + B*u + (-h)*u^2 +/- zcl*v  (log2 domain)
    tC2[idx]  = LOG2E_F * cn - h * In * In;
    tB[idx]   = 2.f * h * In;
    tZcl[idx] = LOG2E_F * 2.f * G / sn2;
  }
}

__global__ __launch_bounds__(256) void bimm_main_kernel(
    const float* __restrict__ u, const float* __restrict__ v,
    const float* __restrict__ ws_ro, float* __restrict__ partials,
    int M, int N, int n_int, int n_phases, int ncomp) {
  __shared__ __align__(16) float smem[SMEM_DW];
  __shared__ float wsum[8];
  const int tid  = threadIdx.x;
  const int lane = tid & 31;
  const bool lo  = (lane < 16);

  // ---- CDNA5 async bulk copy global -> LDS (ASYNCcnt path) ----
  {
    const int nvec = (64 + 3 * n_int * N) >> 2;
    for (int i = tid; i < nvec; i += (int)blockDim.x) {
      uint32_t lds_addr = (uint32_t)(uintptr_t)(&smem[i * 4]);
      uint64_t gaddr    = (uint64_t)(uintptr_t)(ws_ro + i * 4);
      asm volatile("global_load_async_to_lds_b128 %0, %1, off"
                   :: "v"(lds_addr), "v"(gaddr) : "memory");
    }
    asm volatile("s_wait_asynccnt 0" ::: "memory");
  }
  __syncthreads();

  const float inv_sn     = smem[0];
  const float inv_sigeff = smem[1];
  const float inv_sn2    = smem[2];
  const float K_v        = smem[3];
  const float A_int      = smem[4];
  const float logNinv    = smem[5];
  const float Cc         = smem[14];
  const float* tC2  = smem + 64;
  const float* tB   = smem + 64 + n_int * N;
  const float* tZcl = smem + 64 + 2 * n_int * N;
  const float* pYbase = lo ? tB : tZcl;        // per-lane coeff source for K1/K3

  float acc = 0.f;
  const int stride = (int)(gridDim.x * blockDim.x);
  for (int base = (int)(blockIdx.x * blockDim.x); base < M; base += stride) {
    const int  m     = base + tid;
    const bool valid = (m < M);
    const int  mc    = valid ? m : (M - 1);
    __builtin_prefetch(u + m + stride, 0, 0);  // global_prefetch_b8, speculative
    __builtin_prefetch(v + m + stride, 0, 0);
    const float um = u[mc];
    const float vm = v[mc];
    const float logv  = flog(vm);
    const float basev = K_v + logv - vm * vm * inv_sn2;

    // B-matrix (4x16 features x points) for the wave's two 16-point tiles.
    // Layout: VGPR0 = {lanes0-15: K0=1, lanes16-31: K2=u^2}, VGPR1 = {K1=u, K3=v}.
    float bxx[2], byy[2];
    #pragma unroll
    for (int t = 0; t < 2; t++) {
      const int src = t * 16 + (lane & 15);
      const float tu = __shfl(um, src, 32);
      const float tv = __shfl(vm, src, 32);
      bxx[t] = lo ? 1.0f : tu * tu;
      byy[t] = lo ? tu   : tv;
    }

    float lp[16];
    // interior components (cheap closed form)
    float t2 = vm * inv_sigeff; t2 = t2 * t2;
    const float int_base = A_int + 2.f * logv - t2;
    for (int k = 0; k < n_phases; k++) {
      const float du = (um - smem[6 + k]) * inv_sn;
      lp[k] = int_base - 0.5f * du * du;
    }

    // interface components: exponent args via V_WMMA_F32_16X16X4_F32,
    // D[n][m] = c2_n + B_n*u_m + Cc*u_m^2 (+/-) zcl_n*v_m  (log2 domain)
    for (int p = 0; p < n_int; p++) {
      const float* pX = tC2 + p * N;
      const float* pY = pYbase + p * N;
      float sDiff[2];
      #pragma unroll
      for (int t = 0; t < 2; t++) {
        v8f accP = {0.f,0.f,0.f,0.f,0.f,0.f,0.f,0.f};
        v8f accM = {0.f,0.f,0.f,0.f,0.f,0.f,0.f,0.f};
        v2f bb; bb.x = bxx[t]; bb.y = byy[t];
        #pragma unroll
        for (int q = 0; q < 8; q++) {
          const int iA = q * 16 + (lane & 15);
          // A-matrix (16 samples x 4 coeffs): lanes0-15 {K0=c2,K1=B}, lanes16-31 {K2=Cc,K3=+/-zcl}
          const float axl = pX[iA];
          const float ay  = pY[iA];
          v2f aP; aP.x = lo ? axl : Cc; aP.y = ay;
          v2f aM; aM.x = aP.x;          aM.y = lo ? ay : -ay;
          v8f z8 = {0.f,0.f,0.f,0.f,0.f,0.f,0.f,0.f};
          v8f dP = __builtin_amdgcn_wmma_f32_16x16x4_f32(false, aP, false, bb, (short)0, z8, false, false);
          v8f dM = __builtin_amdgcn_wmma_f32_16x16x4_f32(false, aM, false, bb, (short)0, z8, false, false);
          #pragma unroll
          for (int j = 0; j < 8; j++) {
            accP[j] += fexp2(dP[j]);
            accM[j] += fexp2(dM[j]);
          }
        }
        // reduce over the 16 sample-rows: 8 row-VGPRs + other lane half
        float tP = sum8(accP), tM = sum8(accM);
        tP += __shfl_xor(tP, 16, 32);
        tM += __shfl_xor(tM, 16, 32);
        sDiff[t] = tP - tM;              // sum_n exp(arg+) - exp(arg-) for col = lane%16
      }
      const float sOwn = lo ? sDiff[0] : sDiff[1];   // this lane's own m
      lp[n_phases + p] = flog(sOwn) + basev + logNinv;
    }

    // 10-way logsumexp with log_w
    float fmx = -1e30f;
    for (int c = 0; c < ncomp; c++) fmx = fmaxf(fmx, smem[16 + c] + lp[c]);
    float fs = 0.f;
    for (int c = 0; c < ncomp; c++) fs += fexp2(LOG2E_F * (smem[16 + c] + lp[c] - fmx));
    const float val = fmx + flog(fs);
    acc += valid ? val : 0.f;
  }

  // deterministic block reduction: wave32 via WMMA, then serial over <=8 waves
  const float wtot = wave_reduce32(acc);
  if ((tid & 31) == 0) wsum[tid >> 5] = wtot;
  __syncthreads();
  if (tid == 0) {
    float b = 0.f;
    const int nw = (int)blockDim.x >> 5;
    for (int w = 0; w < nw; w++) b += wsum[w];
    partials[blockIdx.x] = b;
  }
}

__global__ void bimm_finish_kernel(const float* __restrict__ partials, int nparts,
                                   float* __restrict__ out, int M) {
  __shared__ float sred[256];
  const int tid = threadIdx.x;
  float s = 0.f;
  for (int i = tid; i < nparts; i += (int)blockDim.x) s += partials[i];
  sred[tid] = s;
  __syncthreads();
  for (int off = 128; off; off >>= 1) {
    if (tid < off) sred[tid] += sred[tid + off];
    __syncthreads();
  }
  if (tid == 0) out[0] = -sred[0] / (float)M;
}

extern "C" void kernel_launch(void* const* d_in, const int* in_sizes, int n_in,
                              void* d_out, int out_size, void* d_ws, size_t ws_size,
                              hipStream_t stream) {
  const float* u   = (const float*)d_in[0];
  const float* v   = (const float*)d_in[1];
  const float* eps = (const float*)d_in[2];
  const float* I   = (const float*)d_in[3];
  const float* W   = (const float*)d_in[4];
  const float* sb  = (const float*)d_in[5];
  const float* sn  = (const float*)d_in[6];
  const float* dp  = (const float*)d_in[7];
  const float* rp  = (const float*)d_in[8];

  const int M        = in_sizes[0];
  const int n_phases = in_sizes[3];          // 4
  const int ncomp    = in_sizes[4];          // 10
  const int n_int    = ncomp - n_phases;     // 6
  const int N        = in_sizes[2] / n_int;  // 128

  float* ws       = (float*)d_ws;
  float* partials = ws + PART_OFF;

  int gridB = (M + 255) / 256;
  if (gridB > 2048) gridB = 2048;

  bimm_setup_kernel<<<1, 256, 0, stream>>>(eps, I, W, sb, sn, dp, rp, ws,
                                           N, n_int, n_phases, ncomp);
  bimm_main_kernel<<<gridB, 256, 0, stream>>>(u, v, ws, partials,
                                              M, N, n_int, n_phases, ncomp);
  bimm_finish_kernel<<<1, 256, 0, stream>>>(partials, gridB, (float*)d_out, M);
}